// MambaGNN_67989332295928
// MI455X (gfx1250) — compile-verified
//
#include <hip/hip_runtime.h>
#include <hip/hip_bf16.h>

typedef float v2f __attribute__((ext_vector_type(2)));
typedef float v8f __attribute__((ext_vector_type(8)));

__device__ __forceinline__ float silu_f(float v) { return v / (1.0f + __expf(-v)); }
__device__ __forceinline__ float softplus_f(float v) {
  return (v > 20.0f) ? v : log1pf(__expf(v));
}

// Adjacency mask A0 rows as 19-bit masks (bit m set => A0[n][m]==1).
__constant__ unsigned kMask[19] = {
  0x5000Bu, 0x40007u, 0x40006u, 0x50109u, 0x40130u, 0x40030u, 0x409C0u,
  0x400C0u, 0x40158u, 0x44E00u, 0x40600u, 0x48A40u, 0x47000u, 0x43000u,
  0x6D200u, 0x7C800u, 0x78009u, 0x7C000u, 0x7FFFFu
};

// One wave32 = one sample. 16 waves (512 threads) per block = 16 samples.
// Phase A: lane = node (GNN + LN + in_proj + conv + x_proj)
// Phase B: lane = (d,s)  (selective scan, 1 exp per (n,d,s))
// Phase C: lane = node (gate, out_proj, residual -> Xo in LDS)
// Phase D: wave 0 does fc1 GEMM tile (16 samples x 18 outs, K=144) with
//          V_WMMA_F32_16X16X4_F32 (36 K-steps x 2 N-tiles = 72 wmma).
__global__ __launch_bounds__(512)
void mamba_gnn_fused(
    const float* __restrict__ x,        const float* __restrict__ gcn_w,
    const float* __restrict__ gcn_b,    const float* __restrict__ ln_w,
    const float* __restrict__ ln_b,     const float* __restrict__ in_proj_w,
    const float* __restrict__ conv_w,   const float* __restrict__ conv_b,
    const float* __restrict__ x_proj_w, const float* __restrict__ dt_proj_w,
    const float* __restrict__ dt_proj_b,const float* __restrict__ A_log,
    const float* __restrict__ Dvec,     const float* __restrict__ out_proj_w,
    const float* __restrict__ fc1_w,    const float* __restrict__ fc1_b,
    float* __restrict__ out)
{
  __shared__ float s_u [16][18][8];
  __shared__ float s_dt[16][18][8];
  __shared__ float s_Bm[16][18][4];
  __shared__ float s_Cm[16][18][4];
  __shared__ float s_y [16][18][8];
  __shared__ float s_Xo[16][144];

  const int lane = threadIdx.x & 31;
  const int w    = threadIdx.x >> 5;          // sample slot within block
  const int samp = blockIdx.x * 16 + w;       // global sample
  const int n    = lane;                      // node id in phases A/C

  // ---------------- Phase A : lane = node ----------------
  // X = concat(x, ones) transposed -> X[n][d], n in [0,19)
  float Xr[8];
#pragma unroll
  for (int d = 0; d < 8; ++d) {
    float v = 0.0f;
    if (n < 18)       v = x[(size_t)samp * 144 + d * 18 + n];
    else if (n == 18) v = 1.0f;
    Xr[d] = v;
  }
  // row L2-normalize
  float ss = 0.0f;
#pragma unroll
  for (int d = 0; d < 8; ++d) ss += Xr[d] * Xr[d];
  const float invn = (n < 19) ? (1.0f / sqrtf(ss)) : 0.0f;
  float X1r[8];
#pragma unroll
  for (int d = 0; d < 8; ++d) X1r[d] = Xr[d] * invn;

  // masked cosine similarity row + degree
  const unsigned mrow = (n < 19) ? kMask[n] : 0u;
  float wrow[19];
  float deg = 0.0f;
#pragma unroll
  for (int m = 0; m < 19; ++m) {
    float dot = 0.0f;
#pragma unroll
    for (int d = 0; d < 8; ++d) dot += X1r[d] * __shfl(X1r[d], m);
    const float wm = ((mrow >> m) & 1u) ? dot : 0.0f;
    wrow[m] = wm;
    deg += wm;
  }
  const float dinv = (deg > 0.0f) ? (1.0f / sqrtf(deg)) : 0.0f;

  // G[n][:] = X[n][:] @ gcn_w^T   (fully lane-local)
  float Gr[8];
#pragma unroll
  for (int dd = 0; dd < 8; ++dd) {
    float acc = 0.0f;
#pragma unroll
    for (int k = 0; k < 8; ++k) acc += Xr[k] * gcn_w[dd * 8 + k];
    Gr[dd] = acc;
  }

  // H[n] = dinv[n] * sum_m wrow[m]*dinv[m]*G[m] + gcn_b   (X4 residual = H)
  float Hr[8];
#pragma unroll
  for (int d = 0; d < 8; ++d) Hr[d] = 0.0f;
#pragma unroll
  for (int m = 0; m < 19; ++m) {
    const float wm = wrow[m] * __shfl(dinv, m);
#pragma unroll
    for (int d = 0; d < 8; ++d) Hr[d] += wm * __shfl(Gr[d], m);
  }
#pragma unroll
  for (int d = 0; d < 8; ++d) Hr[d] = dinv * Hr[d] + gcn_b[d];

  // LayerNorm over (18 nodes x 8 dims), wave butterfly
  float lsum = 0.0f, lsq = 0.0f;
  if (n < 18) {
#pragma unroll
    for (int d = 0; d < 8; ++d) { lsum += Hr[d]; lsq += Hr[d] * Hr[d]; }
  }
#pragma unroll
  for (int off = 16; off >= 1; off >>= 1) {
    lsum += __shfl_xor(lsum, off);
    lsq  += __shfl_xor(lsq,  off);
  }
  const float mu   = lsum * (1.0f / 144.0f);
  const float var  = lsq  * (1.0f / 144.0f) - mu * mu;
  const float rstd = 1.0f / sqrtf(var + 1e-5f);
  const int  nsafe = (n < 18) ? n : 0;
  float Xln[8];
#pragma unroll
  for (int d = 0; d < 8; ++d)
    Xln[d] = (Hr[d] - mu) * rstd * ln_w[nsafe * 8 + d] + ln_b[nsafe * 8 + d];

  // in_proj -> xm, z
  float xmv[8], zv[8];
#pragma unroll
  for (int j = 0; j < 8; ++j) {
    float a0 = 0.0f, a1 = 0.0f;
#pragma unroll
    for (int k = 0; k < 8; ++k) {
      a0 += Xln[k] * in_proj_w[j * 8 + k];
      a1 += Xln[k] * in_proj_w[(j + 8) * 8 + k];
    }
    xmv[j] = a0; zv[j] = a1;
  }

  // causal depth-2 conv along nodes + SiLU
  float uv[8];
#pragma unroll
  for (int d = 0; d < 8; ++d) {
    float xs = __shfl_up(xmv[d], 1);
    if (n == 0) xs = 0.0f;
    uv[d] = silu_f(xs * conv_w[d * 2 + 0] + xmv[d] * conv_w[d * 2 + 1] + conv_b[d]);
  }

  // x_proj -> (dt_raw, B, C)
  float xdbl[9];
#pragma unroll
  for (int j = 0; j < 9; ++j) {
    float a = 0.0f;
#pragma unroll
    for (int d = 0; d < 8; ++d) a += uv[d] * x_proj_w[j * 8 + d];
    xdbl[j] = a;
  }
  if (n < 18) {
#pragma unroll
    for (int d = 0; d < 8; ++d) {
      s_u [w][n][d] = uv[d];
      s_dt[w][n][d] = softplus_f(xdbl[0] * dt_proj_w[d] + dt_proj_b[d]);
    }
#pragma unroll
    for (int s2 = 0; s2 < 4; ++s2) {
      s_Bm[w][n][s2] = xdbl[1 + s2];
      s_Cm[w][n][s2] = xdbl[5 + s2];
    }
  }
  __syncthreads();

  // ---------------- Phase B : lane = (d, s) scan ----------------
  {
    const int d  = lane >> 2;
    const int s2 = lane & 3;
    const float Am = -__expf(A_log[lane]);   // A_log[d][s] at d*4+s == lane
    float h = 0.0f;
#pragma unroll
    for (int t = 0; t < 18; ++t) {
      const float dtv = s_dt[w][t][d];
      const float uu  = s_u [w][t][d];
      const float bb  = s_Bm[w][t][s2];
      const float cc  = s_Cm[w][t][s2];
      h = __expf(dtv * Am) * h + dtv * bb * uu;
      float p = h * cc;
      p += __shfl_xor(p, 1);
      p += __shfl_xor(p, 2);
      if (s2 == 0) s_y[w][t][d] = p;     // y[t][d] = sum_s h*C
    }
  }
  __syncthreads();

  // ---------------- Phase C : lane = node, gate + out_proj + residual ----
  if (n < 18) {
    float yv[8];
#pragma unroll
    for (int d = 0; d < 8; ++d)
      yv[d] = (s_y[w][n][d] + uv[d] * Dvec[d]) * silu_f(zv[d]);
#pragma unroll
    for (int d2 = 0; d2 < 8; ++d2) {
      float acc = 0.0f;
#pragma unroll
      for (int d = 0; d < 8; ++d) acc += yv[d] * out_proj_w[d2 * 8 + d];
      // Xo flat layout: (d, N) -> index d2*18 + n ; residual X4 = Hr
      s_Xo[w][d2 * 18 + n] = Hr[d2] + acc;
    }
  }
  __syncthreads();

  // ---------------- Phase D : wave 0 does fc1 via V_WMMA_F32_16X16X4_F32 --
  if (threadIdx.x < 32) {
    const int mn = lane & 15;            // A: sample row / B: output col
    const int kh = ((lane >> 4) << 1);   // K half-select per 16x4 layout
    const bool ok = (mn < 2);
    const int  r1 = ok ? (16 + mn) : mn; // clamped, always-valid fc1 row
    v8f c0 = {};                         // outputs 0..15
    v8f c1 = {};                         // outputs 16..17 (zero-padded)
#pragma unroll
    for (int k0 = 0; k0 < 144; k0 += 4) {
      v2f a, b0, b1;
      a.x  = s_Xo[mn][k0 + kh];
      a.y  = s_Xo[mn][k0 + kh + 1];
      b0.x = fc1_w[mn * 144 + k0 + kh];
      b0.y = fc1_w[mn * 144 + k0 + kh + 1];
      // Unconditional load from clamped row, then branch-free zero-select
      // (avoids per-element EXEC-mask save/restore sequences).
      const float t0 = fc1_w[r1 * 144 + k0 + kh];
      const float t1 = fc1_w[r1 * 144 + k0 + kh + 1];
      b1.x = ok ? t0 : 0.0f;
      b1.y = ok ? t1 : 0.0f;
      c0 = __builtin_amdgcn_wmma_f32_16x16x4_f32(false, a, false, b0,
                                                 (short)0, c0, false, false);
      c1 = __builtin_amdgcn_wmma_f32_16x16x4_f32(false, a, false, b1,
                                                 (short)0, c1, false, false);
    }
    const int base = blockIdx.x * 16;
#pragma unroll
    for (int r = 0; r < 8; ++r) {
      const int mm = r + ((lane >> 4) << 3);   // C/D layout: M = r + 8*(lane/16)
      out[(size_t)(base + mm) * 18 + mn] = c0[r] + fc1_b[mn];
      if (ok)
        out[(size_t)(base + mm) * 18 + 16 + mn] = c1[r] + fc1_b[16 + mn];
    }
  }
}

extern "C" void kernel_launch(void* const* d_in, const int* in_sizes, int n_in,
                              void* d_out, int out_size, void* d_ws, size_t ws_size,
                              hipStream_t stream) {
  const float* x         = (const float*)d_in[0];
  const float* gcn_w     = (const float*)d_in[1];
  const float* gcn_b     = (const float*)d_in[2];
  const float* ln_w      = (const float*)d_in[3];
  const float* ln_b      = (const float*)d_in[4];
  const float* in_proj_w = (const float*)d_in[5];
  const float* conv_w    = (const float*)d_in[6];
  const float* conv_b    = (const float*)d_in[7];
  const float* x_proj_w  = (const float*)d_in[8];
  const float* dt_proj_w = (const float*)d_in[9];
  const float* dt_proj_b = (const float*)d_in[10];
  const float* A_log     = (const float*)d_in[11];
  const float* Dvec      = (const float*)d_in[12];
  const float* out_proj_w= (const float*)d_in[13];
  const float* fc1_w     = (const float*)d_in[14];
  const float* fc1_b     = (const float*)d_in[15];
  float* out = (float*)d_out;

  const int B = in_sizes[0] / 144;            // 65536
  dim3 grid(B / 16), block(512);
  mamba_gnn_fused<<<grid, block, 0, stream>>>(
      x, gcn_w, gcn_b, ln_w, ln_b, in_proj_w, conv_w, conv_b, x_proj_w,
      dt_proj_w, dt_proj_b, A_log, Dvec, out_proj_w, fc1_w, fc1_b, out);
}